// CustomMLPLayer_27573690040829
// MI455X (gfx1250) — compile-verified
//
#include <hip/hip_runtime.h>
#include <stdint.h>

typedef float v2f __attribute__((ext_vector_type(2)));
typedef float v8f __attribute__((ext_vector_type(8)));

#define KDIM     11008
#define NOUT     4096
#define SEQ      2048
#define KSEL     2201    // int(0.2 * 11008)
#define CORE_NUM 4403    // int(0.4 * 11008)

#define BM 128
#define BN 128
#define BK 32
#define APAD 36   // row stride (floats) for A tile: keeps 8B align, spreads banks
#define BPAD 132  // row stride (floats) for B tile

// ---------------------------------------------------------------------------
// GEMM: Out[s,n] = sum_k X[s,k] * W[n,k]   via V_WMMA_F32_16X16X4_F32 (exact f32)
// ---------------------------------------------------------------------------
__global__ __launch_bounds__(256)
void gemm_f32_wmma(const float* __restrict__ X, const float* __restrict__ W,
                   float* __restrict__ Out)
{
    __shared__ float As[BM][APAD];   // x tile, [m][k]
    __shared__ float Bs[BK][BPAD];   // weight^T tile, [k][n]

    const int bm   = blockIdx.y * BM;
    const int bn   = blockIdx.x * BN;
    const int t    = threadIdx.x;
    const int wave = t >> 5;
    const int lane = t & 31;
    const int half = lane >> 4;      // 0 or 1
    const int l16  = lane & 15;
    const int wm   = (wave & 3) * 32;   // wave's M offset (2 subtiles)
    const int wn   = (wave >> 2) * 64;  // wave's N offset (4 subtiles)
    const int lr   = t >> 3;            // loader row 0..31
    const int lk   = (t & 7) << 2;      // loader k offset 0,4,..,28

    v8f acc[2][4];
#pragma unroll
    for (int i = 0; i < 2; ++i)
#pragma unroll
        for (int j = 0; j < 4; ++j)
            acc[i][j] = (v8f){0.f,0.f,0.f,0.f,0.f,0.f,0.f,0.f};

    for (int k0 = 0; k0 < KDIM; k0 += BK) {
        // ---- stage A (x) and B (weight, transposed) tiles into LDS ----
#pragma unroll
        for (int i = 0; i < 4; ++i) {
            float4 va = *reinterpret_cast<const float4*>(
                X + (size_t)(bm + lr + 32*i) * KDIM + k0 + lk);
            *reinterpret_cast<float4*>(&As[lr + 32*i][lk]) = va;
            float4 vb = *reinterpret_cast<const float4*>(
                W + (size_t)(bn + lr + 32*i) * KDIM + k0 + lk);
            Bs[lk + 0][lr + 32*i] = vb.x;
            Bs[lk + 1][lr + 32*i] = vb.y;
            Bs[lk + 2][lr + 32*i] = vb.z;
            Bs[lk + 3][lr + 32*i] = vb.w;
        }
        __syncthreads();

        // ---- compute: 8 k-steps of 4, 8 WMMAs each ----
#pragma unroll
        for (int kk = 0; kk < BK; kk += 4) {
            // A frag (16x4 f32): lane<16 holds {K=kk,kk+1}, lane>=16 {kk+2,kk+3}
            v2f a[2];
#pragma unroll
            for (int ms = 0; ms < 2; ++ms)
                a[ms] = *reinterpret_cast<const v2f*>(
                    &As[wm + ms*16 + l16][kk + 2*half]);
            // B frag (4x16 f32): v0 = row kk (+2 for hi half), v1 = next row
            v2f b[4];
#pragma unroll
            for (int ns = 0; ns < 4; ++ns) {
                const int col = wn + ns*16 + l16;
                const int kr  = kk + 2*half;
                v2f bb;
                bb.x = Bs[kr][col];
                bb.y = Bs[kr + 1][col];
                b[ns] = bb;
            }
#pragma unroll
            for (int ms = 0; ms < 2; ++ms)
#pragma unroll
                for (int ns = 0; ns < 4; ++ns)
                    acc[ms][ns] = __builtin_amdgcn_wmma_f32_16x16x4_f32(
                        false, a[ms], false, b[ns],
                        (short)0, acc[ms][ns], false, false);
        }
        __syncthreads();
    }

    // C/D layout: VGPR r -> M = r + 8*half, N = l16
#pragma unroll
    for (int ms = 0; ms < 2; ++ms)
#pragma unroll
        for (int ns = 0; ns < 4; ++ns) {
            const int rbase = bm + wm + ms*16 + 8*half;
            const int col   = bn + wn + ns*16 + l16;
#pragma unroll
            for (int r = 0; r < 8; ++r)
                Out[(size_t)(rbase + r) * NOUT + col] = acc[ms][ns][r];
        }
}

// ---------------------------------------------------------------------------
// Per-token top-KSEL membership -> atomic frequency counts.
// Exact radix-select of the k-th largest key; ties at the boundary are taken
// lowest-index-first (matches jax.lax.top_k).
// ---------------------------------------------------------------------------
__global__ __launch_bounds__(256)
void topk_count(const float* __restrict__ X, int* __restrict__ counts)
{
    __shared__ unsigned int keys[KDIM];
    __shared__ int hist[256];
    __shared__ int eqlist[1024];
    __shared__ unsigned int sh_prefix;
    __shared__ int sh_r, sh_eq, sh_nlist;

    const int t = threadIdx.x;
    const float* row = X + (size_t)blockIdx.x * KDIM;

    if (t == 0) { sh_prefix = 0u; sh_r = KSEL; sh_eq = 0; sh_nlist = 0; }
    for (int i = t; i < KDIM; i += 256) {
        unsigned int ub = __float_as_uint(row[i]);
        // order-preserving transform: larger float -> larger uint
        keys[i] = ub ^ (unsigned int)(((int)ub >> 31) | 0x80000000);
    }
    __syncthreads();

    // 4-pass MSB radix select of the KSEL-th largest key
    for (int pass = 0; pass < 4; ++pass) {
        const int shift = 24 - 8 * pass;
        hist[t] = 0;
        __syncthreads();
        const unsigned int pfx  = sh_prefix;
        const unsigned int mask = (pass == 0) ? 0u : (0xFFFFFFFFu << (shift + 8));
        for (int i = t; i < KDIM; i += 256) {
            const unsigned int kv = keys[i];
            if ((kv & mask) == pfx) atomicAdd(&hist[(kv >> shift) & 255], 1);
        }
        __syncthreads();
        if (t == 0) {
            const int rr = sh_r;
            int acc2 = 0;
            for (int d = 255; d >= 0; --d) {
                const int c = hist[d];
                if (acc2 + c >= rr) {
                    sh_r      = rr - acc2;                      // still needed among == T
                    sh_prefix = pfx | ((unsigned int)d << shift);
                    sh_eq     = c;                              // count of == T (so far)
                    break;
                }
                acc2 += c;
            }
        }
        __syncthreads();
    }

    const unsigned int T  = sh_prefix;  // exact k-th largest key
    const int req         = sh_r;       // equals to take (lowest indices first)
    const int cnt_eq      = sh_eq;      // total equals
    const bool take_all   = (cnt_eq == req);

    for (int i = t; i < KDIM; i += 256) {
        const unsigned int kv = keys[i];
        if (kv > T) {
            atomicAdd(&counts[i], 1);
        } else if (kv == T) {
            if (take_all) {
                atomicAdd(&counts[i], 1);
            } else {
                const int p = atomicAdd(&sh_nlist, 1);
                if (p < 1024) eqlist[p] = i;
            }
        }
    }
    __syncthreads();

    if (!take_all) {
        const int nl = sh_nlist;
        if (nl <= 1024) {
            // take the req smallest indices among the ties
            for (int p = t; p < nl; p += 256) {
                const int idx = eqlist[p];
                int rnk = 0;
                for (int q = 0; q < nl; ++q) rnk += (eqlist[q] < idx);
                if (rnk < req) atomicAdd(&counts[idx], 1);
            }
        } else if (t == 0) {  // pathological tie storm: serial, index order
            int need = req;
            for (int i = 0; i < KDIM && need > 0; ++i)
                if (keys[i] == T) { atomicAdd(&counts[i], 1); --need; }
        }
    }
}

__global__ void zero_counts(int* __restrict__ counts)
{
    const int j = blockIdx.x * 256 + threadIdx.x;
    if (j < KDIM) counts[j] = 0;
}

// ---------------------------------------------------------------------------
// core_idx = top-CORE_NUM of counts, descending, ties -> lower index first.
// Composite key (count<<14)|(0x3FFF - j) is unique -> rank = #keys greater.
// ---------------------------------------------------------------------------
__global__ __launch_bounds__(256)
void rank_select(const int* __restrict__ counts, int* __restrict__ core_idx)
{
    __shared__ int sk[KDIM];
    const int t = threadIdx.x;
    for (int i = t; i < KDIM; i += 256)
        sk[i] = (counts[i] << 14) | (0x3FFF - i);
    __syncthreads();

    const int j = blockIdx.x * 256 + t;
    if (j >= KDIM) return;
    const int kj = sk[j];
    int rank = 0;
    for (int i = 0; i < KDIM; ++i) rank += (sk[i] > kj);
    if (rank < CORE_NUM) core_idx[rank] = j;
}

// ---------------------------------------------------------------------------
// filtered_W[m, c] = W[m, core_idx[c]]
// ---------------------------------------------------------------------------
__global__ __launch_bounds__(256)
void gather_w(const float* __restrict__ W, const int* __restrict__ core_idx,
              float* __restrict__ outF)
{
    const long long total = (long long)NOUT * CORE_NUM;
    const long long tid = (long long)blockIdx.x * 256 + threadIdx.x;
    if (tid >= total) return;
    const int c = (int)(tid % CORE_NUM);
    const int m = (int)(tid / CORE_NUM);
    outF[tid] = W[(size_t)m * KDIM + core_idx[c]];
}

// ---------------------------------------------------------------------------
extern "C" void kernel_launch(void* const* d_in, const int* in_sizes, int n_in,
                              void* d_out, int out_size, void* d_ws, size_t ws_size,
                              hipStream_t stream)
{
    (void)in_sizes; (void)n_in; (void)out_size; (void)ws_size;

    const float* x = (const float*)d_in[0];   // [1, 2048, 11008] f32
    const float* w = (const float*)d_in[1];   // [4096, 11008] f32

    float* out        = (float*)d_out;
    float* true_value = out;                          // [2048, 4096]
    float* filtered   = out + (size_t)SEQ * NOUT;     // [4096, 4403]

    int* counts = (int*)d_ws;          // KDIM ints
    int* core   = counts + KDIM;       // CORE_NUM ints

    // GEMM (independent of the selection pipeline)
    dim3 gg(NOUT / BN, SEQ / BM);
    gemm_f32_wmma<<<gg, 256, 0, stream>>>(x, w, true_value);

    // Core-neuron selection pipeline
    zero_counts<<<(KDIM + 255) / 256, 256, 0, stream>>>(counts);
    topk_count<<<SEQ, 256, 0, stream>>>(x, counts);
    rank_select<<<(KDIM + 255) / 256, 256, 0, stream>>>(counts, core);

    const long long total = (long long)NOUT * CORE_NUM;
    gather_w<<<(unsigned)((total + 255) / 256), 256, 0, stream>>>(w, core, filtered);
}